// TwoPhrase_2465311228258
// MI455X (gfx1250) — compile-verified
//
#include <hip/hip_runtime.h>

#define MARGIN   0.2472f
#define SENT_NUM 16384
#define ANN_NUM  8192

typedef __attribute__((ext_vector_type(2))) float v2f;
typedef __attribute__((ext_vector_type(4))) float v4f;
typedef __attribute__((ext_vector_type(8))) float v8f;

// Exact f32 wave-wide sum using V_WMMA_F32_16X16X4_F32.
// A (16x4): lane L<16 supplies (M=L,K=0) in a[0]; lane L>=16 supplies (M=L-16,K=2).
// a[1] = 0 fills K=1/K=3. B (4x16) = all ones. =>  D[m][n] = p_m + p_{m+16}.
// Lane L<16 holds D[0..7][L] in c[0..7]; lane L>=16 holds D[8..15][L-16].
// Sum of c[0..7] gives half the row-sums; shfl_xor(16) combines halves.
__device__ __forceinline__ float wave_reduce_wmma(float p) {
    v2f a; a[0] = p;    a[1] = 0.0f;
    v2f b; b[0] = 1.0f; b[1] = 1.0f;
    v8f c = {};
    c = __builtin_amdgcn_wmma_f32_16x16x4_f32(
            /*neg_a=*/false, a, /*neg_b=*/false, b,
            /*c_mod=*/(short)0, c, /*reuse_a=*/false, /*reuse_b=*/false);
    float s = ((c[0] + c[1]) + (c[2] + c[3])) + ((c[4] + c[5]) + (c[6] + c[7]));
    s += __shfl_xor(s, 16, 32);
    return s;  // full 32-lane sum, valid in every lane
}

// 256-thread (8-wave) block reduction; result valid on thread 0.
__device__ __forceinline__ float block_reduce(float p, float* lds) {
    float w = wave_reduce_wmma(p);
    const int wave = threadIdx.x >> 5;
    const int lane = threadIdx.x & 31;
    if (lane == 0) lds[wave] = w;
    __syncthreads();
    float total = 0.0f;
    if (threadIdx.x == 0) {
#pragma unroll
        for (int i = 0; i < 8; ++i) total += lds[i];
    }
    return total;
}

// One block per row: stream 32 KB with NT b128 loads, hinge, reduce.
// Diagonal (j == pann_i) always contributes exactly MARGIN, subtract it once.
__global__ __launch_bounds__(256) void hinge_row_kernel(
        const float* __restrict__ att,
        const int*   __restrict__ pann,
        float*       __restrict__ partial) {
    __shared__ float lds[8];
    const int    row  = blockIdx.x;
    const size_t base = (size_t)row * (size_t)ANN_NUM;

    const int   idx = pann[row];                 // uniform per block
    const float pos = att[base + (size_t)idx];   // uniform per block
    const float mp  = MARGIN - pos;

    const v4f* rp = (const v4f*)(att + base);    // 2048 float4 per row
    float acc = 0.0f;
#pragma unroll
    for (int k = 0; k < 8; ++k) {
        v4f v = __builtin_nontemporal_load(rp + (threadIdx.x + k * 256));
        acc += fmaxf(v[0] + mp, 0.0f);
        acc += fmaxf(v[1] + mp, 0.0f);
        acc += fmaxf(v[2] + mp, 0.0f);
        acc += fmaxf(v[3] + mp, 0.0f);
    }

    float total = block_reduce(acc, lds);
    if (threadIdx.x == 0) partial[row] = total - MARGIN;
}

// Deterministic final reduction of the 16384 per-row partials.
__global__ __launch_bounds__(256) void final_reduce_kernel(
        const float* __restrict__ partial,
        float*       __restrict__ out) {
    __shared__ float lds[8];
    float acc = 0.0f;
#pragma unroll 8
    for (int k = 0; k < SENT_NUM / 256; ++k)
        acc += partial[threadIdx.x + k * 256];

    float total = block_reduce(acc, lds);
    if (threadIdx.x == 0)
        out[0] = total / ((float)SENT_NUM * (float)ANN_NUM);
}

extern "C" void kernel_launch(void* const* d_in, const int* in_sizes, int n_in,
                              void* d_out, int out_size, void* d_ws, size_t ws_size,
                              hipStream_t stream) {
    const float* att  = (const float*)d_in[0];
    const int*   pann = (const int*)d_in[1];
    float*       part = (float*)d_ws;            // 16384 floats = 64 KB scratch

    hinge_row_kernel<<<SENT_NUM, 256, 0, stream>>>(att, pann, part);
    final_reduce_kernel<<<1, 256, 0, stream>>>(part, (float*)d_out);
}